// GAEconv_12025908429198
// MI455X (gfx1250) — compile-verified
//
#include <hip/hip_runtime.h>
#include <hip/hip_bf16.h>
#include <cstdint>

// ---------------------------------------------------------------------------
// 2-layer GCN on gfx1250 (MI455X, wave32).
//  - GEMMs via V_WMMA_F32_16X16X4_F32 (exact fp32 path). One wave owns a full
//    16-row output stripe (FOUT/16 accumulators) so the A matrix streams from
//    HBM exactly once (X = 102 MB is the dominant HBM stream @ 23.3 TB/s).
//  - GEMM epilogue fuses the accumulator init: stores raw H (scatter-gather
//    source) AND  H*dinv^2 + bias  (self-loop + bias) in one pass.
//  - Edge propagate via L2-resident fp32 hardware atomics (buffers 12-26 MB
//    << 192 MB L2); F/4 consecutive lanes cooperate per edge (coalesced
//    float4 gathers, broadcast index loads).
//  - ReLU fused into GEMM2's A-operand loads.
// ---------------------------------------------------------------------------

typedef __attribute__((ext_vector_type(2))) float v2f;
typedef __attribute__((ext_vector_type(8))) float v8f;

#define F_IN 256
#define HID  32
#define DIM  64

// ---------------- degree / norm ----------------

__global__ void k_deg_init(float* __restrict__ deg, int n) {
    int i = blockIdx.x * blockDim.x + threadIdx.x;
    if (i < n) deg[i] = 1.0f;                     // self-loop contribution
}

__global__ void k_deg_edges(const int* __restrict__ col, float* __restrict__ deg, int nE) {
    int e = blockIdx.x * blockDim.x + threadIdx.x;
    if (e < nE) unsafeAtomicAdd(&deg[col[e]], 1.0f);
}

__global__ void k_dinv(float* __restrict__ deg, int n) {
    int i = blockIdx.x * blockDim.x + threadIdx.x;
    if (i < n) deg[i] = 1.0f / sqrtf(deg[i]);     // deg >= 1 always (self loops)
}

// ---- WMMA GEMM + fused init:
//   Craw [nrows x FOUT] = act(A)[nrows x K] @ B[K x FOUT]
//   Cinit[nrows x FOUT] = Craw * dinv[row]^2 + bias[col]
// One wave computes a full 16-row stripe (TN = FOUT/16 accumulators); the A
// fragment is loaded once per k-step and reused across all N-tiles.
// f32 WMMA frag layout (ISA 7.12.2): lane<16 -> kg=0, lane>=16 -> kg=1;
//   A vgpr0/1 = A[M=lane&15][k0+2kg], A[..][k0+2kg+1]  (contiguous float2)
//   B vgpr0/1 = B[k0+2kg][n], B[k0+2kg+1][n]
//   C/D vgpr j = row 8*kg+j, col n.
// nrows % 16 == 0 here (100000 = 6250*16) so EXEC stays all-1s around WMMA.

template<int K, int FOUT, bool RELU_A>
__global__ __launch_bounds__(128) void k_gemm_fused(const float* __restrict__ A,
                                                    const float* __restrict__ B,
                                                    const float* __restrict__ dinv,
                                                    const float* __restrict__ bias,
                                                    float* __restrict__ Craw,
                                                    float* __restrict__ Cinit,
                                                    int nrows) {
    constexpr int TN = FOUT / 16;
    const int wave  = blockIdx.x * (blockDim.x >> 5) + (threadIdx.x >> 5);
    const int tileM = wave;                       // one wave per 16-row stripe
    if (tileM * 16 >= nrows) return;              // wave-uniform exit

    const int lane = threadIdx.x & 31;
    const int n    = lane & 15;
    const int kg   = lane >> 4;                   // 0 or 1
    const int m    = tileM * 16 + n;              // A-row owned by this lane

    const float* arow = A + (size_t)m * K + 2 * kg;
    const float* bcol = B + n;

    v8f acc[TN];
    #pragma unroll
    for (int t = 0; t < TN; ++t) acc[t] = {};

    #pragma unroll 4
    for (int k0 = 0; k0 < K; k0 += 4) {
        float2 af = *(const float2*)(arow + k0);            // A[m][k0+2kg .. +1]
        if (RELU_A) { af.x = fmaxf(af.x, 0.0f); af.y = fmaxf(af.y, 0.0f); }
        v2f av; av.x = af.x; av.y = af.y;
        #pragma unroll
        for (int t = 0; t < TN; ++t) {                      // B is KB-sized: L0 hits
            float b0 = bcol[(size_t)(k0 + 2 * kg)     * FOUT + t * 16];
            float b1 = bcol[(size_t)(k0 + 2 * kg + 1) * FOUT + t * 16];
            v2f bv; bv.x = b0; bv.y = b1;
            acc[t] = __builtin_amdgcn_wmma_f32_16x16x4_f32(
                         /*neg_a=*/false, av, /*neg_b=*/false, bv,
                         /*c_mod=*/(short)0, acc[t],
                         /*reuse_a=*/false, /*reuse_b=*/false);
        }
    }

    // fused epilogue: self-loop weight per owned row (broadcast loads, cached)
    float w[8];
    #pragma unroll
    for (int j = 0; j < 8; ++j) {
        float d = dinv[tileM * 16 + 8 * kg + j];
        w[j] = d * d;                              // norm of the self edge
    }

    #pragma unroll
    for (int t = 0; t < TN; ++t) {
        float bs = bias[t * 16 + n];
        float* cr = Craw  + ((size_t)(tileM * 16 + 8 * kg)) * FOUT + t * 16 + n;
        float* ci = Cinit + ((size_t)(tileM * 16 + 8 * kg)) * FOUT + t * 16 + n;
        #pragma unroll
        for (int j = 0; j < 8; ++j) {
            float v = acc[t][j];
            cr[(size_t)j * FOUT] = v;
            ci[(size_t)j * FOUT] = fmaf(v, w[j], bs);
        }
    }
}

// ---------------- edge scatter: out[col] += h[row] * dinv[row]*dinv[col] -----

template<int F>
__global__ void k_scatter(const int* __restrict__ row, const int* __restrict__ col,
                          const float* __restrict__ dinv, const float* __restrict__ h,
                          float* __restrict__ out, int nE) {
    const unsigned CH = F / 4;
    unsigned t = blockIdx.x * blockDim.x + threadIdx.x;
    unsigned e = t / CH, c = t % CH;
    if (e >= (unsigned)nE) return;
    int r = row[e];
    int d = col[e];
    float w = dinv[r] * dinv[d];
    float4 v = *(const float4*)(h + (size_t)r * F + 4 * c);
    float* o = out + (size_t)d * F + 4 * c;
    unsafeAtomicAdd(o + 0, v.x * w);
    unsafeAtomicAdd(o + 1, v.y * w);
    unsafeAtomicAdd(o + 2, v.z * w);
    unsafeAtomicAdd(o + 3, v.w * w);
}

// ---------------------------------------------------------------------------

static inline unsigned cdiv(long long a, long long b) { return (unsigned)((a + b - 1) / b); }

extern "C" void kernel_launch(void* const* d_in, const int* in_sizes, int n_in,
                              void* d_out, int out_size, void* d_ws, size_t ws_size,
                              hipStream_t stream) {
    const float* x   = (const float*)d_in[0];   // [N, 256]
    const int*   ei  = (const int*)  d_in[1];   // [2, E]
    const float* W1  = (const float*)d_in[2];   // [256, 32]
    const float* b1  = (const float*)d_in[3];   // [32]
    const float* W2  = (const float*)d_in[4];   // [32, 64]
    const float* b2  = (const float*)d_in[5];   // [64]
    float* outp      = (float*)d_out;           // [N, 64]

    const int N = in_sizes[0] / F_IN;
    const int E = in_sizes[1] / 2;
    const int* row = ei;        // edge_index[0] = src
    const int* col = ei + E;    // edge_index[1] = dst

    // workspace carve-up (16B aligned: N % 4 == 0)
    float* ws   = (float*)d_ws;
    float* dinv = ws;                               // N
    float* hbuf = dinv + N;                         // N*32  raw X@W1 (gather src)
    float* out1 = hbuf + (size_t)N * HID;           // N*32  layer-1 accumulator / H1
    float* h2   = out1 + (size_t)N * HID;           // N*64  raw relu(H1)@W2 (gather src)

    const int T = 256;

    // 1) degree + rsqrt norm
    k_deg_init <<<cdiv(N, T), T, 0, stream>>>(dinv, N);
    k_deg_edges<<<cdiv(E, T), T, 0, stream>>>(col, dinv, E);
    k_dinv     <<<cdiv(N, T), T, 0, stream>>>(dinv, N);

    // 2) GEMM1 (+fused init): hbuf = X@W1 ; out1 = hbuf*dinv^2 + b1
    k_gemm_fused<F_IN, HID, false><<<cdiv(cdiv(N, 16), 4), 128, 0, stream>>>(
        x, W1, dinv, b1, hbuf, out1, N);
    // 3) scatter edges into out1
    k_scatter<HID><<<cdiv((long long)E * (HID / 4), T), T, 0, stream>>>(
        row, col, dinv, hbuf, out1, E);

    // 4) GEMM2 (+fused init, ReLU on A): h2 = relu(out1)@W2 ; d_out = h2*dinv^2 + b2
    k_gemm_fused<HID, DIM, true><<<cdiv(cdiv(N, 16), 4), 128, 0, stream>>>(
        out1, W2, dinv, b2, h2, outp, N);
    // 5) scatter edges into d_out
    k_scatter<DIM><<<cdiv((long long)E * (DIM / 4), T), T, 0, stream>>>(
        row, col, dinv, h2, outp, E);
}